// GraphGCN_58153857188393
// MI455X (gfx1250) — compile-verified
//
#include <hip/hip_runtime.h>

#define N_NODES 100000
#define N_EDGES 1600000
#define DIM 64
#define NEG_SLOPE 0.01f

typedef __attribute__((ext_vector_type(2))) float v2f;
typedef __attribute__((ext_vector_type(8))) float v8f;

// ---------------------------------------------------------------------------
// Zero the aggregation buffer (float4 vectorized).
// ---------------------------------------------------------------------------
__global__ __launch_bounds__(256) void zero_kernel(float4* __restrict__ p, long n4) {
    long i = (long)blockIdx.x * blockDim.x + threadIdx.x;
    if (i < n4) p[i] = make_float4(0.f, 0.f, 0.f, 0.f);
}

// ---------------------------------------------------------------------------
// Edge scatter: agg[dst[e]][d] += x[src[e]][d] * (w[e] if WEIGHTED).
// tid -> (e = tid>>6, d = tid&63): each wave32 covers one edge and 32
// contiguous dims -> 128B coalesced gather + contiguous atomic burst; edge
// index loads are wave-uniform.
// ---------------------------------------------------------------------------
template <bool WEIGHTED>
__global__ __launch_bounds__(256) void scatter_kernel(
    const float* __restrict__ xin, const int* __restrict__ src,
    const int* __restrict__ dst, const float* __restrict__ ew,
    float* __restrict__ agg) {
    long tid = (long)blockIdx.x * blockDim.x + threadIdx.x;
    const long total = (long)N_EDGES * DIM;
    if (tid >= total) return;
    int  d = (int)(tid & (DIM - 1));
    long e = tid >> 6;
    int  s = src[e];
    int  t = dst[e];
    float v = xin[(long)s * DIM + d];
    if (WEIGHTED) v *= ew[e];
    unsafeAtomicAdd(&agg[(long)t * DIM + d], v);
}

// ---------------------------------------------------------------------------
// Fused GraphConv GEMM epilogue:
//   out = leakyrelu(agg @ Wrel^T + brel + xroot @ Wroot^T)
// One wave per 16x16 output tile (V_WMMA_F32_16X16X4_F32, K accumulated
// fp32-exact), 4 waves/block = one 16-row x 64-col strip.
// If FUSED: strip goes to LDS and the final Linear(64->1) dot is applied,
// writing out[N]; else the strip is written to out[N,64].
// ---------------------------------------------------------------------------
template <bool FUSED>
__global__ __launch_bounds__(128) void gemm_kernel(
    const float* __restrict__ agg,    // [N,64]
    const float* __restrict__ xroot,  // [N,64]
    const float* __restrict__ Wrel,   // [64,64] row-major [out][in]
    const float* __restrict__ brel,   // [64]
    const float* __restrict__ Wroot,  // [64,64]
    const float* __restrict__ Wlin,   // [64]   (FUSED only)
    const float* __restrict__ blin,   // [1]    (FUSED only)
    float* __restrict__ out)          // FUSED ? [N] : [N,64]
{
    __shared__ float strip[16 * 64];

    const int wave = threadIdx.x >> 5;
    const int lane = threadIdx.x & 31;
    const int hi   = lane >> 4;     // 0: K pair {k,k+1}, 1: {k+2,k+3}
    const int lm   = lane & 15;     // A: row M;  B: col N (row of W)
    const int m0   = blockIdx.x * 16;
    const int n0   = wave * 16;
    const int koff = hi * 2;

    v8f c = {};

    // C += agg_tile @ Wrel^T  (B[k][n] = Wrel[n][k] -> lane reads Wrel row)
    const float* arow = agg  + (long)(m0 + lm) * DIM + koff;
    const float* brow = Wrel + (long)(n0 + lm) * DIM + koff;
#pragma unroll
    for (int k0 = 0; k0 < DIM; k0 += 4) {
        v2f a = *(const v2f*)(arow + k0);
        v2f b = *(const v2f*)(brow + k0);
        c = __builtin_amdgcn_wmma_f32_16x16x4_f32(false, a, false, b,
                                                  (short)0, c, false, false);
    }
    // C += xroot_tile @ Wroot^T
    const float* arow2 = xroot + (long)(m0 + lm) * DIM + koff;
    const float* brow2 = Wroot + (long)(n0 + lm) * DIM + koff;
#pragma unroll
    for (int k0 = 0; k0 < DIM; k0 += 4) {
        v2f a = *(const v2f*)(arow2 + k0);
        v2f b = *(const v2f*)(brow2 + k0);
        c = __builtin_amdgcn_wmma_f32_16x16x4_f32(false, a, false, b,
                                                  (short)0, c, false, false);
    }

    // Epilogue: bias (per output col) + leaky relu.
    const float bias = brel[n0 + lm];
#pragma unroll
    for (int v = 0; v < 8; ++v) {
        float h = c[v] + bias;
        h = (h >= 0.f) ? h : NEG_SLOPE * h;
        const int row = v + hi * 8;               // C layout: VGPR v -> M
        if (FUSED) {
            strip[row * 64 + n0 + lm] = h;
        } else {
            out[(long)(m0 + row) * DIM + n0 + lm] = h;
        }
    }

    if (FUSED) {
        __syncthreads();
        if (threadIdx.x < 16) {
            const int r = threadIdx.x;
            float sum = blin[0];
#pragma unroll
            for (int j = 0; j < 64; ++j) {
                const int k = (j + r * 4) & 63;   // skew start: avoid LDS bank conflicts
                sum += strip[r * 64 + k] * Wlin[k];
            }
            out[m0 + r] = sum;
        }
    }
}

// ---------------------------------------------------------------------------
extern "C" void kernel_launch(void* const* d_in, const int* in_sizes, int n_in,
                              void* d_out, int out_size, void* d_ws, size_t ws_size,
                              hipStream_t stream) {
    const float* x      = (const float*)d_in[0];
    const int*   ei     = (const int*)  d_in[1];   // [2,E] row-major
    const float* ew     = (const float*)d_in[2];
    const float* Wrel1  = (const float*)d_in[3];
    const float* brel1  = (const float*)d_in[4];
    const float* Wroot1 = (const float*)d_in[5];
    const float* Wrel2  = (const float*)d_in[6];
    const float* brel2  = (const float*)d_in[7];
    const float* Wroot2 = (const float*)d_in[8];
    const float* Wlin   = (const float*)d_in[9];
    const float* blin   = (const float*)d_in[10];
    float*       out    = (float*)d_out;

    const int* src = ei;            // edge_index[0]
    const int* dst = ei + N_EDGES;  // edge_index[1]

    float* bufA = (float*)d_ws;                  // agg (layer 1, then layer 2)
    float* bufB = bufA + (long)N_NODES * DIM;    // h

    const long nElems = (long)N_NODES * DIM;     // 6.4M
    const long n4     = nElems / 4;
    const int  zgrid  = (int)((n4 + 255) / 256);
    const long sTot   = (long)N_EDGES * DIM;     // 102.4M
    const int  sgrid  = (int)((sTot + 255) / 256);
    const int  strips = N_NODES / 16;            // 6250

    // ---- layer 1 ----
    zero_kernel<<<zgrid, 256, 0, stream>>>((float4*)bufA, n4);
    scatter_kernel<true><<<sgrid, 256, 0, stream>>>(x, src, dst, ew, bufA);
    gemm_kernel<false><<<strips, 128, 0, stream>>>(bufA, x, Wrel1, brel1, Wroot1,
                                                   nullptr, nullptr, bufB);
    // ---- layer 2 + head ----
    zero_kernel<<<zgrid, 256, 0, stream>>>((float4*)bufA, n4);
    scatter_kernel<false><<<sgrid, 256, 0, stream>>>(bufB, src, dst, nullptr, bufA);
    gemm_kernel<true><<<strips, 128, 0, stream>>>(bufA, bufB, Wrel2, brel2, Wroot2,
                                                  Wlin, blin, out);
}